// DifferentiableHungarianLoss_16879221473581
// MI455X (gfx1250) — compile-verified
//
#include <hip/hip_runtime.h>
#include <hip/hip_bf16.h>
#include <math.h>

#define NN 512

typedef float v2f __attribute__((ext_vector_type(2)));
typedef float v8f __attribute__((ext_vector_type(8)));

// ---------------------------------------------------------------------------
// Kernel 1: per-row squared norms of A (latent[0]) and B (latent[1])
// ---------------------------------------------------------------------------
__global__ __launch_bounds__(256) void norms_kernel(const float* __restrict__ latent,
                                                    float* __restrict__ na,
                                                    float* __restrict__ nb) {
    int i = blockIdx.x * blockDim.x + threadIdx.x;
    if (i >= NN) return;
    const float* a = latent + (size_t)i * NN;
    const float* b = latent + (size_t)NN * NN + (size_t)i * NN;
    float sa = 0.f, sb = 0.f;
    for (int k = 0; k < NN; ++k) {
        sa += a[k] * a[k];
        sb += b[k] * b[k];
    }
    na[i] = sa;
    nb[i] = sb;
}

// ---------------------------------------------------------------------------
// Kernel 2: cost[i][j] = sqrt(max(|a_i|^2 + |b_j|^2 - 2 a_i.b_j, 1e-12))
// One wave32 computes one 16x16 tile via V_WMMA_F32_16X16X4_F32, K-loop of 128.
// A-frag (16x4 f32): lanes 0-15 -> M=lane, K={k,k+1}; lanes 16-31 -> K={k+2,k+3}.
// B-frag (4x16 f32): mirrored with N across lanes.
// C/D (16x16 f32, 8 VGPRs): VGPR r -> row = r + 8*(lane>=16), col = lane&15.
// ---------------------------------------------------------------------------
__global__ __launch_bounds__(256) void cost_kernel(const float* __restrict__ latent,
                                                   const float* __restrict__ na,
                                                   const float* __restrict__ nb,
                                                   float* __restrict__ cost) {
    const float* A = latent;
    const float* B = latent + (size_t)NN * NN;

    const int lane = threadIdx.x & 31;
    const int wid  = threadIdx.x >> 5;
    const int tile = blockIdx.x * 8 + wid;      // 1024 tiles total
    const int tm   = (tile >> 5) * 16;          // row base
    const int tn   = (tile & 31) * 16;          // col base
    const int hi   = lane >> 4;                 // 0 or 1
    const int l    = lane & 15;

    const float* arow = A + (size_t)(tm + l) * NN + 2 * hi;
    const float* brow = B + (size_t)(tn + l) * NN + 2 * hi;

    v8f acc = {};
#if defined(__gfx1250__)
    for (int k = 0; k < NN; k += 4) {
        v2f av = *(const v2f*)(arow + k);
        v2f bv = *(const v2f*)(brow + k);
        // D = A x B + C   (16x16x4, f32)
        acc = __builtin_amdgcn_wmma_f32_16x16x4_f32(
            /*neg_a=*/false, av, /*neg_b=*/false, bv,
            /*c_mod=*/(short)0, acc, /*reuse_a=*/false, /*reuse_b=*/false);
    }
#endif
    const float nbv = nb[tn + l];
    for (int r = 0; r < 8; ++r) {
        const int row = tm + r + 8 * hi;
        const int col = tn + l;
        float d2 = na[row] + nbv - 2.0f * acc[r];
        cost[(size_t)row * NN + col] = sqrtf(fmaxf(d2, 1e-12f));
    }
}

// ---------------------------------------------------------------------------
// Kernel 3: Jonker-Volgenant Hungarian (f64 duals, matching the numpy ref),
// single 512-thread workgroup (16 wave32s). Thread t owns column j = t+1.
// Then fused finalize: loss + row/col indices.
// ---------------------------------------------------------------------------
__global__ __launch_bounds__(512) void hungarian_kernel(const float* __restrict__ cost,
                                                        float* __restrict__ out) {
    __shared__ double u[NN + 1], v[NN + 1], minv[NN + 1];
    __shared__ int p[NN + 1], way[NN + 1];
    __shared__ unsigned char used[NN + 1];
    __shared__ double redV[16];
    __shared__ int    redI[16];
    __shared__ int    j1_sh;
    __shared__ double delta_sh;
    __shared__ int    colind[NN];
    __shared__ double rA[16], rB[16];

    const int tid  = threadIdx.x;
    const int lane = tid & 31;
    const int wid  = tid >> 5;
    const double DINF = __builtin_inf();
    const int j = tid + 1;                       // column owned by this thread

    u[j] = 0.0; v[j] = 0.0; p[j] = 0;
    if (tid == 0) { u[0] = 0.0; v[0] = 0.0; p[0] = 0; }
    __syncthreads();

    for (int i = 1; i <= NN; ++i) {
        minv[j] = DINF; used[j] = 0; way[j] = 0;
        if (tid == 0) { p[0] = i; used[0] = 0; }
        __syncthreads();

        int j0 = 0;
        while (true) {
            if (tid == 0) used[j0] = 1;
            __syncthreads();
            const int i0 = p[j0];

            double val = DINF;
            if (!used[j]) {
                double cur = (double)cost[(size_t)(i0 - 1) * NN + (j - 1)] - u[i0] - v[j];
                if (cur < minv[j]) { minv[j] = cur; way[j] = j0; }
                val = minv[j];
            }
            int jj = j;
            // intra-wave argmin (keep lower lane on ties -> lower column index)
            for (int off = 16; off > 0; off >>= 1) {
                double ov = __shfl_down(val, off, 32);
                int    oj = __shfl_down(jj,  off, 32);
                if (ov < val) { val = ov; jj = oj; }
            }
            if (lane == 0) { redV[wid] = val; redI[wid] = jj; }
            __syncthreads();
            if (tid == 0) {
                double bv = redV[0]; int bj = redI[0];
                for (int w = 1; w < 16; ++w)
                    if (redV[w] < bv) { bv = redV[w]; bj = redI[w]; }
                delta_sh = bv; j1_sh = bj;
            }
            __syncthreads();
            const double delta = delta_sh;
            const int j1 = j1_sh;

            // dual updates (used columns map to distinct rows -> no races)
            if (used[j]) { u[p[j]] += delta; v[j] -= delta; }
            else         { minv[j] -= delta; }
            if (tid == 0) { u[p[0]] += delta; v[0] -= delta; }  // j0==0 always used
            __syncthreads();

            j0 = j1;
            if (p[j0] == 0) break;
        }
        // augment along the alternating path (serial, thread 0)
        if (tid == 0) {
            int jc = j0;
            while (jc != 0) { int jn = way[jc]; p[jc] = p[jn]; jc = jn; }
        }
        __syncthreads();
    }

    // ---- finalize: col_ind[row] and loss ----
    colind[p[j] - 1] = j - 1;                    // p is a permutation of 1..N
    __syncthreads();

    double tr = (double)cost[(size_t)tid * NN + tid];
    double pr = (double)cost[(size_t)tid * NN + colind[tid]];
    for (int off = 16; off > 0; off >>= 1) {
        tr += __shfl_down(tr, off, 32);
        pr += __shfl_down(pr, off, 32);
    }
    if (lane == 0) { rA[wid] = tr; rB[wid] = pr; }
    __syncthreads();
    if (tid == 0) {
        double sa = 0.0, sb = 0.0;
        for (int w = 0; w < 16; ++w) { sa += rA[w]; sb += rB[w]; }
        out[0] = (float)(fabs(sb - sa) / (double)NN);
    }
    out[1 + tid]      = (float)tid;              // row_ind
    out[1 + NN + tid] = (float)colind[tid];      // col_ind
}

// ---------------------------------------------------------------------------
// Launch
// ---------------------------------------------------------------------------
extern "C" void kernel_launch(void* const* d_in, const int* in_sizes, int n_in,
                              void* d_out, int out_size, void* d_ws, size_t ws_size,
                              hipStream_t stream) {
    (void)in_sizes; (void)n_in; (void)out_size; (void)ws_size;
    const float* latent = (const float*)d_in[0];
    float* out  = (float*)d_out;

    float* cost = (float*)d_ws;                  // 512*512 f32 = 1 MB
    float* na   = cost + (size_t)NN * NN;        // 512 f32
    float* nb   = na + NN;                       // 512 f32

    norms_kernel<<<2, 256, 0, stream>>>(latent, na, nb);
    cost_kernel<<<128, 256, 0, stream>>>(latent, na, nb, cost);
    hungarian_kernel<<<1, 512, 0, stream>>>(cost, out);
}